// NPCGCN_1468878815350
// MI455X (gfx1250) — compile-verified
//
#include <hip/hip_runtime.h>

typedef __attribute__((ext_vector_type(2))) float v2f;
typedef __attribute__((ext_vector_type(8))) float v8f;

#define NSRC0 500000
#define NDST0 100000
#define NE0   1000000
#define NDST1 10000
#define NE1   100000
#define NDST2 1024
#define NE2   10240
#define DIM   128
#define NCLS  47

// ---------------- utility kernels ----------------

__global__ void gcn_fill_zero(float* __restrict__ p, long long n) {
  long long i = (long long)blockIdx.x * blockDim.x + threadIdx.x;
  long long s = (long long)gridDim.x * blockDim.x;
  for (; i < n; i += s) p[i] = 0.0f;
}

__global__ void gcn_count_deg(const int* __restrict__ idx, float* __restrict__ deg, int e) {
  int i = blockIdx.x * blockDim.x + threadIdx.x;
  if (i < e) atomicAdd(&deg[idx[i]], 1.0f);
}

__global__ void gcn_deg_to_rsqrt(float* __restrict__ p, int n) {
  int i = blockIdx.x * blockDim.x + threadIdx.x;
  if (i < n) p[i] = rsqrtf(fmaxf(p[i], 1.0f));
}

// out[dst[e]][:] += rows[src[e]][:], D=128; one wave per edge, float4 per lane
__global__ void gcn_edge_scatter(const float* __restrict__ rows,
                                 const int* __restrict__ src,
                                 const int* __restrict__ dst,
                                 float* __restrict__ out, int e) {
  long long t = (long long)blockIdx.x * blockDim.x + threadIdx.x;
  int eidx = (int)(t >> 5);
  int lane = (int)(t & 31);
  if (eidx >= e) return;
  int s = src[eidx];
  int d = dst[eidx];
  const float4 v = *(const float4*)(rows + (long long)s * DIM + lane * 4);
  float* o = out + (long long)d * DIM + lane * 4;
  atomicAdd(o + 0, v.x);
  atomicAdd(o + 1, v.y);
  atomicAdd(o + 2, v.z);
  atomicAdd(o + 3, v.w);
}

// out[i][:] = h[inverse[shuffle[i]]][:] * rs[i]   (one wave per row)
__global__ void gcn_permute_scale(const float* __restrict__ h,
                                  const int* __restrict__ inv,
                                  const int* __restrict__ shuf,
                                  const float* __restrict__ rs,
                                  float* __restrict__ out, int n) {
  long long t = (long long)blockIdx.x * blockDim.x + threadIdx.x;
  int i = (int)(t >> 5);
  int lane = (int)(t & 31);
  if (i >= n) return;
  int p = inv[shuf[i]];
  float s = rs[i];
  float4 v = *(const float4*)(h + (long long)p * DIM + lane * 4);
  v.x *= s; v.y *= s; v.z *= s; v.w *= s;
  *(float4*)(out + (long long)i * DIM + lane * 4) = v;
}

// ---------------- fp32 WMMA GEMM ----------------
// out[M x N] = epilogue( A[M x 128] @ W[128 x N] ), K = 128
// Block covers 64 rows (4 x 16-row WMMA tiles) so one B fragment feeds 4 WMMAs.
// Wave w handles cols [16w, 16w+16); blockDim = 32 * ceil(N/16).
// epilogue: v = (RELU? max(0, .) : .)(acc * rs_pre[m] + bias[n]) * rs_post[m]
#define LDSTR 132   // padded row stride (floats): conflict-free b64 LDS reads
#define MTILES 4
__global__ __launch_bounds__(256)
void gcn_fill_unused() {} // keep symbol section tidy

template <bool PRE, bool POST, bool RELU, bool COLGUARD>
__global__ void gcn_gemm_wmma_f32(const float* __restrict__ A,
                                  const float* __restrict__ W,
                                  const float* __restrict__ bias,
                                  const float* __restrict__ rs_pre,
                                  const float* __restrict__ rs_post,
                                  float* __restrict__ out,
                                  int M, int N) {
  __shared__ float As[MTILES * 16 * LDSTR];
  const int tid = threadIdx.x;
  const int m0  = blockIdx.x * (MTILES * 16);

  // Stage 64x128 A tile into LDS with gfx1250 async global->LDS copies.
  // Each idx moves one float4; LDS row stride padded to 132 floats.
  for (int idx = tid; idx < (MTILES * 16 * DIM) / 4; idx += blockDim.x) {
    int r  = idx >> 5;             // staged row 0..63
    int c4 = (idx & 31) * 4;       // column base
    int rg = m0 + r;
    rg = rg < M ? rg : (M - 1);    // clamp tail rows (results masked at store)
    const float* gp = &A[(long long)rg * DIM + c4];
    unsigned ldsoff = (unsigned)(size_t)(&As[r * LDSTR + c4]);
    asm volatile("global_load_async_to_lds_b128 %0, %1, off"
                 :: "v"(ldsoff), "v"(gp) : "memory");
  }
  asm volatile("s_wait_asynccnt 0" ::: "memory");
  __syncthreads();

  const int wave  = tid >> 5;
  const int lane  = tid & 31;
  const int l15   = lane & 15;
  const int khalf = (lane >> 4) << 1;          // 0 (lanes 0-15) or 2 (lanes 16-31)
  const int col   = wave * 16 + l15;
  const int colc  = (COLGUARD && col >= N) ? 0 : col;  // clamp for loads (N=47 tail)

  v8f acc[MTILES];
  #pragma unroll
  for (int t = 0; t < MTILES; ++t) acc[t] = (v8f){};

  for (int k = 0; k < DIM; k += 4) {
    v2f b;
    const float* bp = &W[(k + khalf) * N + colc];
    b.x = bp[0];
    b.y = bp[N];
    #pragma unroll
    for (int t = 0; t < MTILES; ++t) {
      v2f a;
      const float* ap = &As[(t * 16 + l15) * LDSTR + k + khalf];
      a.x = ap[0];
      a.y = ap[1];
      acc[t] = __builtin_amdgcn_wmma_f32_16x16x4_f32(false, a, false, b,
                                                     (short)0, acc[t],
                                                     false, false);
    }
  }

  // epilogue + store: lane L covers rows mt + (L<16?0:8) + r, col = 16*wave + L%16
  const float bn   = bias[colc];
  const int   roff = (lane >> 4) << 3;         // 0 or 8
  #pragma unroll
  for (int t = 0; t < MTILES; ++t) {
    int mt = m0 + t * 16;
    if (mt >= M) break;                        // uniform guard (M % 16 == 0)
    #pragma unroll
    for (int r = 0; r < 8; ++r) {
      int row = mt + roff + r;
      float v = acc[t][r];
      if (PRE)  v *= rs_pre[row];
      v += bn;
      if (RELU) v = fmaxf(v, 0.0f);
      if (POST) v *= rs_post[row];
      if (!COLGUARD || col < N) out[(long long)row * N + col] = v;
    }
  }
}

// ---------------- launch ----------------

extern "C" void kernel_launch(void* const* d_in, const int* in_sizes, int n_in,
                              void* d_out, int out_size, void* d_ws, size_t ws_size,
                              hipStream_t stream) {
  const float* feats   = (const float*)d_in[0];
  const int*   src0    = (const int*)d_in[1];
  const int*   dst0    = (const int*)d_in[2];
  const int*   src1    = (const int*)d_in[3];
  const int*   dst1    = (const int*)d_in[4];
  const int*   src2    = (const int*)d_in[5];
  const int*   dst2    = (const int*)d_in[6];
  const int*   inv_idx = (const int*)d_in[7];
  const int*   shf_idx = (const int*)d_in[8];
  const float* W0      = (const float*)d_in[9];
  const float* b0      = (const float*)d_in[10];
  const float* W1      = (const float*)d_in[11];
  const float* b1      = (const float*)d_in[12];
  const float* W2      = (const float*)d_in[13];
  const float* b2      = (const float*)d_in[14];
  float* out = (float*)d_out;

  float* ws = (float*)d_ws;
  size_t o = 0;
  float* agg0     = ws + o; o += (size_t)NDST0 * DIM;   // zeroed each call; reused as h1src later
  float* agg1     = ws + o; o += (size_t)NDST1 * DIM;
  float* agg2     = ws + o; o += (size_t)NDST2 * DIM;
  float* rs_out1  = ws + o; o += NDST0;                 // rsqrt(out_deg of src1) over N_DST0
  float* rs_in1   = ws + o; o += NDST1;
  float* rs_out2  = ws + o; o += NDST1;
  float* rs_in2   = ws + o; o += NDST2;
  const long long zero_count = (long long)o;
  float* h0r      = ws + o; o += (size_t)NDST0 * DIM;   // relu(agg0@W0 + b0)
  float* h2src    = ws + o; o += (size_t)NDST1 * DIM;   // layer-2 scaled source rows
  float* h1src    = agg0;                               // reuse (agg0 dead after GEMM0)

  const int B = 256;

  // 1) zero accumulators + degree arrays
  gcn_fill_zero<<<4096, B, 0, stream>>>(ws, zero_count);

  // 2) degrees -> rsqrt(max(deg,1))
  gcn_count_deg<<<(NE1 + B - 1) / B, B, 0, stream>>>(src1, rs_out1, NE1);
  gcn_count_deg<<<(NE1 + B - 1) / B, B, 0, stream>>>(dst1, rs_in1, NE1);
  gcn_count_deg<<<(NE2 + B - 1) / B, B, 0, stream>>>(src2, rs_out2, NE2);
  gcn_count_deg<<<(NE2 + B - 1) / B, B, 0, stream>>>(dst2, rs_in2, NE2);
  {
    const int ndeg = NDST0 + NDST1 + NDST1 + NDST2;     // contiguous region
    gcn_deg_to_rsqrt<<<(ndeg + B - 1) / B, B, 0, stream>>>(rs_out1, ndeg);
  }

  // 3) layer 0: aggregate raw features, then transform (linearity of matmul)
  {
    long long t = (long long)NE0 * 32;
    gcn_edge_scatter<<<(unsigned)((t + B - 1) / B), B, 0, stream>>>(feats, src0, dst0, agg0, NE0);
  }
  gcn_gemm_wmma_f32<false, false, true, false>
      <<<(NDST0 + 63) / 64, 256, 0, stream>>>(agg0, W0, b0, nullptr, nullptr,
                                              h0r, NDST0, DIM);

  // 4) composed permutation + src-norm for layer 1
  {
    long long t = (long long)NDST0 * 32;
    gcn_permute_scale<<<(unsigned)((t + B - 1) / B), B, 0, stream>>>(h0r, inv_idx, shf_idx,
                                                                     rs_out1, h1src, NDST0);
  }

  // 5) layer 1: aggregate, transform, dst-norm + bias + relu + next-layer src-norm
  {
    long long t = (long long)NE1 * 32;
    gcn_edge_scatter<<<(unsigned)((t + B - 1) / B), B, 0, stream>>>(h1src, src1, dst1, agg1, NE1);
  }
  gcn_gemm_wmma_f32<true, true, true, false>
      <<<(NDST1 + 63) / 64, 256, 0, stream>>>(agg1, W1, b1, rs_in1, rs_out2,
                                              h2src, NDST1, DIM);

  // 6) layer 2: aggregate, transform (N=47, padded tiles), dst-norm + bias
  {
    long long t = (long long)NE2 * 32;
    gcn_edge_scatter<<<(unsigned)((t + B - 1) / B), B, 0, stream>>>(h2src, src2, dst2, agg2, NE2);
  }
  gcn_gemm_wmma_f32<true, false, false, true>
      <<<(NDST2 + 63) / 64, 96, 0, stream>>>(agg2, W2, b2, rs_in2, nullptr,
                                             out, NDST2, NCLS);
}